// Correlation_36077725286488
// MI455X (gfx1250) — compile-verified
//
#include <hip/hip_runtime.h>

typedef __attribute__((ext_vector_type(2))) float v2f;
typedef __attribute__((ext_vector_type(4))) float f4;
typedef __attribute__((ext_vector_type(8))) float v8f;

#define LDS_STRIDE 80   // floats per channel row in LDS (64 data + pad; 80%64=16 -> conflict-free halves)
#define X2_OFF 4        // x2 data stored at idx w'+4; cols 0..3 and 68..79 are zero (W-border padding)

// x1,x2: (B=2, C=64, D=64, H=64, W=64) f32.  out: (B, 27, D, H, W) f32.
// One 128-thread workgroup per (b,d,h). Banded-WMMA over W, band = the 3 dz shifts.
__global__ __launch_bounds__(128)
void corr3d_wmma_kernel(const float* __restrict__ x1,
                        const float* __restrict__ x2,
                        float* __restrict__ out) {
    __shared__ float lds1[64 * LDS_STRIDE];       // x1[c][w]  (w at idx w; idx 64..79 zero)
    __shared__ float lds2[3 * 64 * LDS_STRIDE];   // x2 rows dy=0..2: [c][w'] at idx w'+4

    const int bid = blockIdx.x;          // 0 .. 8191
    const int h   = bid & 63;
    const int d   = (bid >> 6) & 63;
    const int b   = bid >> 12;

    const int t    = threadIdx.x;        // 0..127
    const int c    = t >> 1;             // channel this thread copies
    const int half = t & 1;              // which 32-float half of the W row
    const int lane = t & 31;
    const int wid  = t >> 5;             // wave id 0..3
    const int m    = lane & 15;          // WMMA row (A) / column (B/D) index
    const int hi   = lane >> 4;          // lane-half selects K pair {0,1} vs {2,3}

    const f4 zero4 = {0.f, 0.f, 0.f, 0.f};

    // global base for this thread's channel slice (b, c, *, *, half*32)
    const size_t inBase = (size_t)b * 16777216u + (size_t)c * 262144u + (size_t)(half * 32);

    // ---- load x1 row: 64c x 64w, 128-bit copies, zero-pad idx 64..79 ----
    {
        const float* src = x1 + inBase + (size_t)d * 4096u + (size_t)h * 64u;
        float* dst = lds1 + c * LDS_STRIDE + half * 32;
        #pragma unroll
        for (int i = 0; i < 8; ++i)
            *(f4*)(dst + 4 * i) = *(const f4*)(src + 4 * i);
        if (half) {                       // dst+32 .. dst+47  ==  idx 64..79
            #pragma unroll
            for (int i = 8; i < 12; ++i) *(f4*)(dst + 4 * i) = zero4;
        }
    }

    for (int u = 0; u < 3; ++u) {         // dx displacement index (d + u - 1)
        __syncthreads();                  // protect lds2 reuse across u iterations
        const int d2 = d + u - 1;

        // ---- load x2 rows dy=0..2 for this dx (zero row if D/H out of range) ----
        #pragma unroll
        for (int v = 0; v < 3; ++v) {
            const int h2 = h + v - 1;
            float* row = lds2 + (v * 64 + c) * LDS_STRIDE;
            float* dst = row + X2_OFF + half * 32;
            const bool valid = (d2 >= 0) & (d2 < 64) & (h2 >= 0) & (h2 < 64);
            if (valid) {
                const float* src = x2 + inBase + (size_t)d2 * 4096u + (size_t)h2 * 64u;
                #pragma unroll
                for (int i = 0; i < 8; ++i)
                    *(f4*)(dst + 4 * i) = *(const f4*)(src + 4 * i);
            } else {
                #pragma unroll
                for (int i = 0; i < 8; ++i) *(f4*)(dst + 4 * i) = zero4;
            }
            if (half == 0) {
                *(f4*)(row) = zero4;                      // w' = -4..-1
            } else {
                *(f4*)(row + 68) = zero4;                 // w' = 64..75
                *(f4*)(row + 72) = zero4;
                *(f4*)(row + 76) = zero4;
            }
        }
        __syncthreads();

        // ---- 15 products = 5 W-tiles x 3 dy, distributed over 4 waves ----
        for (int p = wid; p < 15; p += 4) {
            const int tile = p / 3;           // 0..4
            const int v    = p - 3 * tile;    // dy index 0..2
            const int w0   = 14 * tile;       // overlapping tiles, stride 14
            const float* Brow = lds2 + v * 64 * LDS_STRIDE;

            v8f acc = {};
            #pragma unroll
            for (int k = 0; k < 16; ++k) {    // 16 x K=4 accumulation over C=64
                const int c0 = 4 * k + 2 * hi;
                v2f av, bv;
                // A[m][k] = x1[c0+k'][w0+m]      (16x4 f32 A layout)
                av.x = lds1[c0 * LDS_STRIDE + w0 + m];
                av.y = lds1[(c0 + 1) * LDS_STRIDE + w0 + m];
                // B[k][n] = x2[c0+k'][w0-1+n]    (n = lane%16, stored at w'+4)
                bv.x = Brow[c0 * LDS_STRIDE + (w0 + 3 + m)];
                bv.y = Brow[(c0 + 1) * LDS_STRIDE + (w0 + 3 + m)];
                acc = __builtin_amdgcn_wmma_f32_16x16x4_f32(
                        false, av, false, bv, (short)0, acc, false, false);
            }

            // ---- band extraction: P[mm][n] with s = n - mm in [0,2] is out[w0+mm, dz=s-1] ----
            const int n = m;
            const int dd_base = u * 9 + v * 3;
            #pragma unroll
            for (int j = 0; j < 8; ++j) {
                const int mm = 8 * hi + j;        // D-matrix row owned by this VGPR
                const int s  = n - mm;            // dz index 0..2
                const int w  = w0 + mm;
                if (mm <= 13 && s >= 0 && s <= 2 && w < 64) {
                    const size_t o = (size_t)(b * 27 + dd_base + s) * 262144u
                                   + (size_t)d * 4096u + (size_t)h * 64u + (size_t)w;
                    out[o] = acc[j] * 0.015625f;  // mean over C=64
                }
            }
        }
    }
}

extern "C" void kernel_launch(void* const* d_in, const int* in_sizes, int n_in,
                              void* d_out, int out_size, void* d_ws, size_t ws_size,
                              hipStream_t stream) {
    const float* x1 = (const float*)d_in[0];
    const float* x2 = (const float*)d_in[1];
    float* out = (float*)d_out;
    (void)in_sizes; (void)n_in; (void)out_size; (void)d_ws; (void)ws_size;
    dim3 grid(2 * 64 * 64);   // one workgroup per (b, d, h)
    dim3 block(128);          // 4 waves of 32
    hipLaunchKernelGGL(corr3d_wmma_kernel, grid, block, 0, stream, x1, x2, out);
}